// ImprovedSSIMLoss_74380243632260
// MI455X (gfx1250) — compile-verified
//
#include <hip/hip_runtime.h>

typedef float v2f __attribute__((ext_vector_type(2)));
typedef float v8f __attribute__((ext_vector_type(8)));

// 11-tap gaussian (sigma=1.5), normalized; symmetric.
__constant__ float GW[11] = {
    0.00102838f, 0.00759878f, 0.03600080f, 0.10936080f, 0.21300550f,
    0.26601170f,
    0.21300550f, 0.10936080f, 0.03600080f, 0.00759878f, 0.00102838f};

__device__ __forceinline__ float gwf(int t) {
    return (t >= 0 && t < 11) ? GW[t] : 0.0f;
}

#define C1 0.0001f
#define C2 0.0009f
#define IMG 512
#define OUTD 502            // 512 - 11 + 1
#define TILES_PER_PLANE 1024  // 32x32 tiles of 16x16
#define NPLANES 48
#define WAVES_PER_BLOCK 4
#define NBLOCKS ((NPLANES * TILES_PER_PLANE) / WAVES_PER_BLOCK)  // 12288
#define LDSF_PER_WAVE 3120  // 784 (XR) + 896 (EX) + 896 (YT) + 544 (HT)

__global__ __launch_bounds__(128) void ssim_tile_kernel(
    const float* __restrict__ x, const float* __restrict__ y,
    float* __restrict__ partial) {
    __shared__ float smem[WAVES_PER_BLOCK * LDSF_PER_WAVE + 128];

    const int w    = threadIdx.x >> 5;
    const int lane = threadIdx.x & 31;
    const int half = lane >> 4;   // 0: lanes 0-15, 1: lanes 16-31
    const int lm   = lane & 15;   // M (A/D) or N (B/D) index

    float* XR = smem + w * LDSF_PER_WAVE;  // 28x28 raw x halo
    float* EX = XR + 784;                  // 32x28 edge-enhanced x (zero-pad)
    float* YT = EX + 896;                  // 32x28 y tile (zero-pad)
    float* HT = YT + 896;                  // 32x17 horizontal-blur scratch
    float* sred = smem + WAVES_PER_BLOCK * LDSF_PER_WAVE;

    const int tileIdx = blockIdx.x * WAVES_PER_BLOCK + w;
    const int plane   = tileIdx >> 10;
    const int t       = tileIdx & 1023;
    const int or0     = (t >> 5) << 4;
    const int oc0     = (t & 31) << 4;
    const int pbase   = plane * (IMG * IMG);

    // ---- phase 1: zero-init padded tiles, load raw halo + y ----
    for (int i = lane; i < 896; i += 32) { EX[i] = 0.0f; YT[i] = 0.0f; }
    for (int i = lane; i < 784; i += 32) {  // 28x28 halo of y_true
        int r = i / 28, c = i - r * 28;
        int gr = or0 - 1 + r, gc = oc0 - 1 + c;
        float v = 0.0f;
        if (gr >= 0 && gr < IMG && gc >= 0 && gc < IMG)
            v = x[pbase + gr * IMG + gc];
        XR[i] = v;
    }
    for (int i = lane; i < 676; i += 32) {  // 26x26 of y_pred
        int r = i / 26, c = i - r * 26;
        int gr = or0 + r, gc = oc0 + c;
        float v = 0.0f;
        if (gr < IMG && gc < IMG) v = y[pbase + gr * IMG + gc];
        YT[r * 28 + c] = v;
    }
    __syncthreads();

    // ---- phase 2: edge-enhance (4-neighbour laplacian, SAME zero-pad) ----
    for (int i = lane; i < 676; i += 32) {
        int r = i / 26, c = i - r * 26;
        float ctr = XR[(r + 1) * 28 + (c + 1)];
        float up  = XR[r * 28 + (c + 1)];
        float dn  = XR[(r + 2) * 28 + (c + 1)];
        float lf  = XR[(r + 1) * 28 + c];
        float rt  = XR[(r + 1) * 28 + (c + 2)];
        EX[r * 28 + c] = up + dn + lf + rt - 4.0f * ctr;
    }
    __syncthreads();

    // ---- gaussian band fragments (shared by horiz-B and vert-A) ----
    // B[k][n] = g[k-n], Gv[m][k] = g[k-m]; per-lane value g[kk - lm]
    v2f gf[7];
#pragma unroll
    for (int s = 0; s < 7; ++s) {
        int kk = 4 * s + 2 * half;
        gf[s][0] = gwf(kk - lm);
        gf[s][1] = gwf(kk + 1 - lm);
    }

    v8f acc[5];
    const v8f vz = {0.f, 0.f, 0.f, 0.f, 0.f, 0.f, 0.f, 0.f};

#pragma unroll
    for (int q = 0; q < 5; ++q) {  // 0:mu1 1:mu2 2:blur(xx) 3:blur(yy) 4:blur(xy)
        __syncthreads();
        // horizontal blur: H[rb+m][n] = sum_k q[rb+m][k] * g[k-n]
#pragma unroll
        for (int rb = 0; rb < 32; rb += 16) {
            v8f h = vz;
            const int row = rb + lm;
#pragma unroll
            for (int s = 0; s < 7; ++s) {
                int kk = 4 * s + 2 * half;
                float e0, e1;
                if (q == 0) {
                    e0 = EX[row * 28 + kk]; e1 = EX[row * 28 + kk + 1];
                } else if (q == 1) {
                    e0 = YT[row * 28 + kk]; e1 = YT[row * 28 + kk + 1];
                } else if (q == 2) {
                    float a0 = EX[row * 28 + kk], a1 = EX[row * 28 + kk + 1];
                    e0 = a0 * a0; e1 = a1 * a1;
                } else if (q == 3) {
                    float a0 = YT[row * 28 + kk], a1 = YT[row * 28 + kk + 1];
                    e0 = a0 * a0; e1 = a1 * a1;
                } else {
                    float a0 = EX[row * 28 + kk], a1 = EX[row * 28 + kk + 1];
                    float b0 = YT[row * 28 + kk], b1 = YT[row * 28 + kk + 1];
                    e0 = a0 * b0; e1 = a1 * b1;
                }
                v2f a; a[0] = e0; a[1] = e1;
                h = __builtin_amdgcn_wmma_f32_16x16x4_f32(
                    false, a, false, gf[s], (short)0, h, false, false);
            }
#pragma unroll
            for (int v = 0; v < 8; ++v) {
                int hr = rb + v + 8 * half;
                HT[hr * 17 + lm] = h[v];
            }
        }
        __syncthreads();
        // vertical blur: out[m][n] = sum_k g[k-m] * H[k][n]
        v8f o = vz;
#pragma unroll
        for (int s = 0; s < 7; ++s) {
            int kk = 4 * s + 2 * half;
            v2f b;
            b[0] = HT[kk * 17 + lm];
            b[1] = HT[(kk + 1) * 17 + lm];
            o = __builtin_amdgcn_wmma_f32_16x16x4_f32(
                false, gf[s], false, b, (short)0, o, false, false);
        }
        acc[q] = o;
    }

    // ---- SSIM map + masked local sum (C/D layout: row=v+8*half, col=lm) ----
    float lsum = 0.0f;
#pragma unroll
    for (int v = 0; v < 8; ++v) {
        float m1 = acc[0][v], m2 = acc[1][v];
        float sxx = acc[2][v] - m1 * m1;
        float syy = acc[3][v] - m2 * m2;
        float sxy = acc[4][v] - m1 * m2;
        float cs = (2.0f * sxy + C2) / (sxx + syy + C2);
        float lu = (2.0f * m1 * m2 + C1) / (m1 * m1 + m2 * m2 + C1);
        int row = or0 + v + 8 * half;
        int col = oc0 + lm;
        if (row < OUTD && col < OUTD) lsum += lu * cs;
    }

    // ---- block tree-reduce -> deterministic partial ----
    sred[threadIdx.x] = lsum;
    __syncthreads();
#pragma unroll
    for (int off = 64; off > 0; off >>= 1) {
        if (threadIdx.x < off) sred[threadIdx.x] += sred[threadIdx.x + off];
        __syncthreads();
    }
    if (threadIdx.x == 0) partial[blockIdx.x] = sred[0];
}

__global__ __launch_bounds__(256) void ssim_finalize_kernel(
    const float* __restrict__ partial, int n, float* __restrict__ out) {
    __shared__ float s[256];
    float a = 0.0f;
    for (int i = threadIdx.x; i < n; i += 256) a += partial[i];
    s[threadIdx.x] = a;
    __syncthreads();
#pragma unroll
    for (int off = 128; off > 0; off >>= 1) {
        if (threadIdx.x < off) s[threadIdx.x] += s[threadIdx.x + off];
        __syncthreads();
    }
    if (threadIdx.x == 0)
        out[0] = 1.0f - s[0] / 12096192.0f;  // 48 * 502 * 502
}

extern "C" void kernel_launch(void* const* d_in, const int* in_sizes, int n_in,
                              void* d_out, int out_size, void* d_ws, size_t ws_size,
                              hipStream_t stream) {
    const float* y_true = (const float*)d_in[0];
    const float* y_pred = (const float*)d_in[1];
    float* out = (float*)d_out;
    float* partial = (float*)d_ws;  // NBLOCKS floats = 48 KB

    ssim_tile_kernel<<<NBLOCKS, 128, 0, stream>>>(y_true, y_pred, partial);
    ssim_finalize_kernel<<<1, 256, 0, stream>>>(partial, NBLOCKS, out);
}